// BiStochastic_28750511079904
// MI455X (gfx1250) — compile-verified
//
#include <hip/hip_runtime.h>
#include <math.h>

// ---------------------------------------------------------------------------
// Log-domain Sinkhorn via potentials on MI455X (gfx1250).
//   log_s^(k) = t - r[i] - c[j],  t = s / TAU
//   row pass:  r[i] = lse_{j<nc}( t[i,j] - c[j] )  two-phase (max, sum-exp),
//              register-resident row tile, wave32 shuffle combine
//   col pass:  c[j] = log( sum_{i<nr} exp(t[i,j] - r[i]) )  -- exponent
//              provably bounded => no stabilization => pure ones x E matmul
//              done on the matrix pipe with V_WMMA_F32_16X16X4_F32
// One block per batch (128 x 1024), potentials (4 KB) in LDS, matrix stays
// L2-resident (128 MB < 192 MB L2) across all 11 passes. Native v_exp/v_log
// keep transcendentals on the TRANS pipe, co-executing with VALU/WMMA.
// ---------------------------------------------------------------------------

typedef float v2f __attribute__((ext_vector_type(2)));
typedef float v8f __attribute__((ext_vector_type(8)));

#define NEGV (-1e30f)

__global__ __launch_bounds__(1024)
void BiStochastic_sinkhorn_kernel(const float* __restrict__ s,
                                  const int* __restrict__ nrows,
                                  const int* __restrict__ ncols,
                                  float* __restrict__ out)
{
    constexpr int N = 512, M = 512;
    const int b = blockIdx.x;
    const float* __restrict__ sb = s + (size_t)b * N * M;
    float* __restrict__ ob = out + (size_t)b * N * M;
    const int nr = nrows[b];
    const int nc = ncols[b];
    const float invTau = 1.0f;  // TAU == 1.0

    __shared__ float r_lds[N];
    __shared__ float c_lds[M];

    const int tid  = threadIdx.x;
    const int lane = tid & 31;
    const int wave = tid >> 5;  // 0..31 (wave32)

    // c^0 = 0
    if (tid < M) c_lds[tid] = 0.0f;
    __syncthreads();

    if (nr > 0 && nc > 0) {
        for (int it = 0; it < 5; ++it) {
            // -------- Row pass: two-phase LSE, row tile in registers -------
            // Lane owns columns {4*lane + 128*k + e}; c is loop-invariant
            // across rows -> preload this lane's 16 c values once.
            float4 cv[4];
            #pragma unroll
            for (int k = 0; k < 4; ++k)
                cv[k] = reinterpret_cast<const float4*>(c_lds)[lane + 32 * k];

            for (int i = wave; i < nr; i += 32) {
                const float4* __restrict__ row4 =
                    reinterpret_cast<const float4*>(sb + i * M);
                float xs[16];
                #pragma unroll
                for (int k = 0; k < 4; ++k) {
                    float4 v = row4[lane + 32 * k];
                    int jb = 4 * lane + 128 * k;
                    xs[4*k+0] = (jb+0 < nc) ? v.x * invTau - cv[k].x : NEGV;
                    xs[4*k+1] = (jb+1 < nc) ? v.y * invTau - cv[k].y : NEGV;
                    xs[4*k+2] = (jb+2 < nc) ? v.z * invTau - cv[k].z : NEGV;
                    xs[4*k+3] = (jb+3 < nc) ? v.w * invTau - cv[k].w : NEGV;
                }
                float m = NEGV;
                #pragma unroll
                for (int t = 0; t < 16; ++t) m = fmaxf(m, xs[t]);
                float a = 0.0f;
                #pragma unroll
                for (int t = 0; t < 16; ++t) a += __expf(xs[t] - m);

                // wave32 cross-lane (m,a) combine
                #pragma unroll
                for (int off = 16; off > 0; off >>= 1) {
                    float mo = __shfl_xor(m, off, 32);
                    float ao = __shfl_xor(a, off, 32);
                    float m2 = fmaxf(m, mo);
                    a = a * __expf(m - m2) + ao * __expf(mo - m2);
                    m = m2;
                }
                if (lane == 0) r_lds[i] = m + __logf(a);
            }
            __syncthreads();

            // -------- Col pass: ones(16x4) x expE(4x16) on matrix pipe -----
            // 32 waves x 16 columns. A==1 makes B's K-slot layout
            // irrelevant; only lane->N mapping matters (lanes 0-15 == N).
            {
                const int jb   = wave << 4;
                const int nsub = lane & 15;
                const int rsel = (lane < 16) ? 0 : 2;
                const int jcol = jb + nsub;

                v8f acc = {0.f, 0.f, 0.f, 0.f, 0.f, 0.f, 0.f, 0.f};
                v2f ones; ones.x = 1.0f; ones.y = 1.0f;

                #pragma unroll 4
                for (int i0 = 0; i0 < nr; i0 += 4) {
                    int r0 = i0 + rsel;
                    int r1 = r0 + 1;
                    int rr0 = (r0 < N - 1) ? r0 : (N - 1);
                    int rr1 = (r1 < N - 1) ? r1 : (N - 1);
                    float x0 = sb[rr0 * M + jcol] * invTau - r_lds[rr0];
                    float x1 = sb[rr1 * M + jcol] * invTau - r_lds[rr1];
                    float e0 = __expf(x0);
                    float e1 = __expf(x1);
                    // branchless tail mask: keep EXEC all-ones for WMMA
                    e0 = (r0 < nr) ? e0 : 0.0f;
                    e1 = (r1 < nr) ? e1 : 0.0f;
                    v2f bmat; bmat.x = e0; bmat.y = e1;
                    acc = __builtin_amdgcn_wmma_f32_16x16x4_f32(
                        false, ones, false, bmat, (short)0, acc, false, false);
                }
                // Every D row carries the same column sums; VGPR0 lanes 0-15
                // hold (M=0, N=lane).
                if (lane < 16) {
                    float cs = acc[0];
                    c_lds[jb + lane] = (cs > 0.0f) ? __logf(cs) : NEGV;
                }
            }
            __syncthreads();
        }
    }

    // -------- Final pass: out = mask ? exp(t - r - c) : 0, float4 I/O ------
    // (harness poisons d_out -> every element must be written)
    {
        const float4* __restrict__ s4 = reinterpret_cast<const float4*>(sb);
        float4* __restrict__ o4 = reinterpret_cast<float4*>(ob);
        for (int idx = tid; idx < N * M / 4; idx += 1024) {
            int i  = idx >> 7;          // 128 float4s per row
            int j0 = (idx & 127) << 2;
            float4 v = s4[idx];
            float ri = r_lds[i];
            float4 c = reinterpret_cast<const float4*>(c_lds)[idx & 127];
            bool rowok = (i < nr);
            float4 o;
            o.x = (rowok && j0+0 < nc) ? __expf(v.x * invTau - ri - c.x) : 0.0f;
            o.y = (rowok && j0+1 < nc) ? __expf(v.y * invTau - ri - c.y) : 0.0f;
            o.z = (rowok && j0+2 < nc) ? __expf(v.z * invTau - ri - c.z) : 0.0f;
            o.w = (rowok && j0+3 < nc) ? __expf(v.w * invTau - ri - c.w) : 0.0f;
            o4[idx] = o;
        }
    }
}

extern "C" void kernel_launch(void* const* d_in, const int* in_sizes, int n_in,
                              void* d_out, int out_size, void* d_ws, size_t ws_size,
                              hipStream_t stream) {
    const float* s     = (const float*)d_in[0];
    const int*   nrows = (const int*)d_in[1];
    const int*   ncols = (const int*)d_in[2];
    float*       out   = (float*)d_out;
    const int B = in_sizes[1];  // 128 batches; one workgroup per batch
    hipLaunchKernelGGL(BiStochastic_sinkhorn_kernel,
                       dim3(B), dim3(1024), 0, stream,
                       s, nrows, ncols, out);
}